// TrackConnectivityComputer_72172630442358
// MI455X (gfx1250) — compile-verified
//
#include <hip/hip_runtime.h>
#include <hip/hip_bf16.h>
#include <stdint.h>

#define NF    50      // node feature count
#define NPORT 6       // ports per node
#define NB    16      // batch
#define NN    1024    // nodes
#define KSTEPS 13     // ceil(50 / 4)

typedef float v2f __attribute__((ext_vector_type(2)));
typedef float v8f __attribute__((ext_vector_type(8)));

__device__ __forceinline__ int imin(int a, int b) { return a < b ? a : b; }

// ---------------------------------------------------------------------------
// Kernel 1: port_activity = node_features @ port_feature_mask via
// V_WMMA_F32_16X16X4_F32, then compress (activity > 0) into a 6-bit mask per
// (node, batch) with wave ballots. Output: bm[node*16 + batch] (16 KB).
// One wave per 16-row tile of the flattened (B*N, F) A matrix.
// All pad handling is clamp-address + cndmask (no divergent branches).
// ---------------------------------------------------------------------------
__global__ __launch_bounds__(128) void port_mask_wmma_kernel(
    const float* __restrict__ nf,   // (B*N, 50) row-major
    const float* __restrict__ pm,   // (50, 6) row-major
    uint8_t* __restrict__ bm)       // (N, 16) byte masks
{
    const int lane  = threadIdx.x & 31;
    const int wave  = (int)((blockIdx.x * blockDim.x + threadIdx.x) >> 5);
    const int row0  = wave * 16;            // first of 16 rows in this tile
    const int m     = lane & 15;            // row-in-tile (A) / column (B)
    const int khalf = (lane >> 4) << 1;     // K sub-offset: 0 or 2
    const int row   = row0 + m;

    // --- Preload all B fragments (padded 52x16, cols 6..15 zero) -----------
    // Unconditional loads from clamped addresses; zero-pad via select.
    const int  colc   = imin(m, NPORT - 1);
    const bool colok  = (m < NPORT);
    v2f bfrag[KSTEPS];
#pragma unroll
    for (int t = 0; t < KSTEPS; ++t) {
        const int kk  = t * 4 + khalf;
        const float bx = pm[imin(kk,     NF - 1) * NPORT + colc];
        const float by = pm[imin(kk + 1, NF - 1) * NPORT + colc];
        bfrag[t].x = (colok && kk     < NF) ? bx : 0.0f;
        bfrag[t].y = (colok && kk + 1 < NF) ? by : 0.0f;
    }

    // --- K loop: A fragment load + WMMA ------------------------------------
    v8f c = {};
#pragma unroll
    for (int t = 0; t < KSTEPS; ++t) {
        const int kk = t * 4 + khalf;
        const float ax = nf[(size_t)row * NF + imin(kk,     NF - 1)];
        const float ay = nf[(size_t)row * NF + imin(kk + 1, NF - 1)];
        v2f a;
        a.x = (kk     < NF) ? ax : 0.0f;
        a.y = (kk + 1 < NF) ? ay : 0.0f;
        // D = A*B + C  (8 args: neg_a, A, neg_b, B, c_mod, C, reuse_a, reuse_b)
        c = __builtin_amdgcn_wmma_f32_16x16x4_f32(
                false, a, false, bfrag[t], (short)0, c, false, false);
    }

    // C/D layout: VGPR r -> lanes 0-15 = (M=r, N=lane), lanes 16-31 = (M=r+8).
    // One ballot of (c[r] > 0) yields row r's port bits in [5:0] and
    // row (r+8)'s port bits in [21:16].
    const int batch = row0 >> 10;           // 64 tiles per batch
    const int n0    = row0 & (NN - 1);
#pragma unroll
    for (int r = 0; r < 8; ++r) {
        unsigned bal = __builtin_amdgcn_ballot_w32(c[r] > 0.0f);
        if (lane == 0) {
            bm[(n0 + r)     * NB + batch] = (uint8_t)(bal & 0x3Fu);
            bm[(n0 + r + 8) * NB + batch] = (uint8_t)((bal >> 16) & 0x3Fu);
        }
    }
}

// ---------------------------------------------------------------------------
// Kernel 2: out[b,i,j] = bit(src[b,i], dc) & bit(dst[b,j], oc) & (dir!=6).
// One block per row i; each thread owns 4 consecutive j's and writes one
// coalesced float4 per batch (16 float4 stores = 256B per thread).
// ---------------------------------------------------------------------------
__global__ __launch_bounds__(256) void connectivity_kernel(
    const int* __restrict__ dir,    // (N, N)
    const uint8_t* __restrict__ bm, // (N, 16)
    float* __restrict__ out)        // (B, N, N)
{
    const int i  = (int)blockIdx.x;
    const int j4 = (int)threadIdx.x * 4;

    const int4  d4   = *(const int4*)(dir + (size_t)i * NN + j4);
    const uint4 srcv = *(const uint4*)(bm + i * NB);

    const int dd[4] = { d4.x, d4.y, d4.z, d4.w };
    unsigned dc[4], oc[4], live[4];
    uint4 dstv[4];
#pragma unroll
    for (int l = 0; l < 4; ++l) {
        const int d = dd[l];
        dc[l]   = (d > 5) ? 5u : (unsigned)d;     // min(d, 5)
        int o   = d + 3; if (o >= 6) o -= 6;      // (d+3) % 6, d <= 6
        oc[l]   = (unsigned)o;
        live[l] = (d != 6) ? 1u : 0u;
        dstv[l] = *(const uint4*)(bm + (j4 + l) * NB);
    }

    const unsigned* sw = (const unsigned*)&srcv;
#pragma unroll
    for (int b = 0; b < NB; ++b) {
        const unsigned sb = (sw[b >> 2] >> ((b & 3) * 8)) & 0xFFu;
        float4 r;
        float* rp = (float*)&r;
#pragma unroll
        for (int l = 0; l < 4; ++l) {
            const unsigned* dw = (const unsigned*)&dstv[l];
            const unsigned db  = (dw[b >> 2] >> ((b & 3) * 8)) & 0xFFu;
            const unsigned bit = (sb >> dc[l]) & (db >> oc[l]) & live[l] & 1u;
            rp[l] = bit ? 1.0f : 0.0f;
        }
        *(float4*)(out + ((size_t)b << 20) + (size_t)i * NN + j4) = r;
    }
}

// ---------------------------------------------------------------------------
extern "C" void kernel_launch(void* const* d_in, const int* in_sizes, int n_in,
                              void* d_out, int out_size, void* d_ws, size_t ws_size,
                              hipStream_t stream) {
    const float* nf  = (const float*)d_in[0];   // node_features (16,1024,50) f32
    const int*   dir = (const int*)d_in[1];     // direction_matrix (1024,1024) i32
    const float* pm  = (const float*)d_in[2];   // port_feature_mask (50,6) f32
    float* out = (float*)d_out;                 // (16,1024,1024) f32
    uint8_t* bm = (uint8_t*)d_ws;               // 1024*16 bytes of scratch

    // 16384 rows / 16 per wave = 1024 waves; 4 waves per 128-thread block.
    port_mask_wmma_kernel<<<dim3(256), dim3(128), 0, stream>>>(nf, pm, bm);
    // One block per row i; 256 threads x 4 j's cover all 1024 columns.
    connectivity_kernel<<<dim3(NN), dim3(256), 0, stream>>>(dir, bm, out);
}